// GCN_10393820856762
// MI455X (gfx1250) — compile-verified
//
#include <hip/hip_runtime.h>
#include <hip/hip_bf16.h>

typedef float v2f __attribute__((ext_vector_type(2)));
typedef float v8f __attribute__((ext_vector_type(8)));

#define HDIM 64

// ---------------------------------------------------------------- utilities
__global__ void zero_kernel(float* __restrict__ p, int n) {
    int i = blockIdx.x * blockDim.x + threadIdx.x;
    int stride = gridDim.x * blockDim.x;
    for (; i < n; i += stride) p[i] = 0.0f;
}

// ------------------------------------------------- edge pass 1: deg / sign / scalar neighbor sum
__global__ void edge_stats_kernel(const int* __restrict__ src, const int* __restrict__ dst,
                                  const float* __restrict__ x,
                                  float* __restrict__ degacc, float* __restrict__ signacc,
                                  float* __restrict__ Sx, int E) {
    int e = blockIdx.x * blockDim.x + threadIdx.x;
    if (e >= E) return;
    int s = src[e], d = dst[e];
    float sg = (s > d) ? 1.0f : ((s < d) ? -1.0f : 0.0f);
    atomicAdd(&degacc[d], 1.0f);
    atomicAdd(&signacc[d], sg);
    atomicAdd(&Sx[d], x[s]);
}

__global__ void node_stats_kernel(const float* __restrict__ degacc, const float* __restrict__ signacc,
                                  float* __restrict__ invdeg, float* __restrict__ gate,
                                  float* __restrict__ ssn, int N) {
    int i = blockIdx.x * blockDim.x + threadIdx.x;
    if (i >= N) return;
    float dg = degacc[i];
    float inv = 1.0f / fmaxf(dg, 1.0f);
    invdeg[i] = inv;
    gate[i]   = (dg > 0.0f) ? 1.0f : 0.0f;
    ssn[i]    = signacc[i] * inv;
}

// ------------------------------------------------- precompute Wd = Wa - Wb for the 6 mid layers
__global__ void wdiff_kernel(const float* __restrict__ Wmid, float* __restrict__ Wd6, int total) {
    int i = blockIdx.x * blockDim.x + threadIdx.x;
    if (i >= total) return;
    int l = i >> 12;          // / 4096
    int rc = i & 4095;
    const float* Wl = Wmid + (size_t)l * 129 * HDIM;
    Wd6[i] = Wl[rc] - Wl[64 * HDIM + rc];
}

// ------------------------------------------------- conv_in: F=1 -> H=64 (factored, no GEMM needed)
__global__ void conv_in_kernel(const float* __restrict__ x, const float* __restrict__ Win,
                               const float* __restrict__ bin,
                               const float* __restrict__ invdeg, const float* __restrict__ gate,
                               const float* __restrict__ ssn, const float* __restrict__ Sx,
                               float* __restrict__ Y, int N) {
    int t = blockIdx.x * blockDim.x + threadIdx.x;
    if (t >= N * HDIM) return;
    int n = t >> 6, h = t & 63;
    float wa = Win[h], wb = Win[HDIM + h], wc = Win[2 * HDIM + h];
    Y[t] = gate[n] * (x[n] * (wa - wb) + bin[h]) + invdeg[n] * Sx[n] * wb + ssn[n] * wc;
}

// ------------------------------------------------- 64-wide neighbor-sum scatter (the O(E) part)
__global__ void scatter_kernel(const int* __restrict__ src, const int* __restrict__ dst,
                               const float* __restrict__ X, float* __restrict__ S, int work) {
    int t = blockIdx.x * blockDim.x + threadIdx.x;
    if (t >= work) return;
    int e = t >> 4;           // edge
    int q = t & 15;           // quad of 4 floats
    int s = src[e], d = dst[e];
    const float4 v = *(const float4*)(X + (size_t)s * HDIM + q * 4);
    float* o = S + (size_t)d * HDIM + q * 4;
    atomicAdd(o + 0, v.x);
    atomicAdd(o + 1, v.y);
    atomicAdd(o + 2, v.z);
    atomicAdd(o + 3, v.w);
}

// ------------------------------------------------- mid conv: per-node dual GEMM via f32 WMMA
// Y = act( gate*(X@Wd + b) + invdeg*(S@Wb) + ssn*Wc [+ resid] ),  act = leaky_relu(0.01)
//
// Weights are staged into LDS pre-swizzled into B-fragment order:
//   sW[((ks*4 + nt)*32 + lane)*2 + {0,1}] = W[ks*4 + kh + {0,1}][nt*16 + (lane&15)]
// with kh = 2*(lane>>4), so each lane's B operand for (ks, nt) is one contiguous
// 8-byte v2f -> a single conflict-free ds_load_b64 straight into an even VGPR pair.
__global__ __launch_bounds__(256) void mid_conv_kernel(
    const float* __restrict__ X, const float* __restrict__ S,
    const float* __restrict__ Wd, const float* __restrict__ Wb,
    const float* __restrict__ Wc, const float* __restrict__ bias,
    const float* __restrict__ gate, const float* __restrict__ invdeg,
    const float* __restrict__ ssn, const float* __restrict__ resid,
    float* __restrict__ Y, int nTiles, int N) {
    __shared__ float sWd[HDIM * HDIM];
    __shared__ float sWb[HDIM * HDIM];
    for (int i = threadIdx.x; i < HDIM * HDIM; i += 256) {
        int ks   = i >> 8;          // k-step          0..15
        int rem  = i & 255;
        int nt   = rem >> 6;        // column tile     0..3
        int lp   = rem & 63;        // lane*2 + half
        int ln   = lp >> 1;
        int half = lp & 1;
        int kh   = (ln >> 4) << 1;
        int srcr = (ks << 2) + kh + half;
        int srcc = (nt << 4) + (ln & 15);
        sWd[i] = Wd[srcr * HDIM + srcc];
        sWb[i] = Wb[srcr * HDIM + srcc];
    }
    __syncthreads();

    int wave = threadIdx.x >> 5;
    int lane = threadIdx.x & 31;
    int tile = blockIdx.x * 8 + wave;
    if (tile >= nTiles) return;          // wave-uniform: EXEC all-ones inside WMMA region

    int node0 = tile * 16;
    int m  = lane & 15;                  // A row / C column index
    int kh = (lane >> 4) << 1;           // upper half-wave handles K+2,K+3
    int row = node0 + m; if (row >= N) row = N - 1;
    const float* Xr = X + (size_t)row * HDIM;
    const float* Sr = S + (size_t)row * HDIM;

    v8f accX[4] = {};
    v8f accS[4] = {};
    int fragBase = lane * 2;

#pragma unroll
    for (int ks = 0; ks < 16; ++ks) {
        int k0 = ks << 2;
        v2f a, s;
        a.x = Xr[k0 + kh];     a.y = Xr[k0 + kh + 1];
        s.x = Sr[k0 + kh];     s.y = Sr[k0 + kh + 1];
#pragma unroll
        for (int nt = 0; nt < 4; ++nt) {
            v2f bd = *(const v2f*)&sWd[((ks << 2) + nt) * 64 + fragBase];
            v2f bb = *(const v2f*)&sWb[((ks << 2) + nt) * 64 + fragBase];
            accX[nt] = __builtin_amdgcn_wmma_f32_16x16x4_f32(
                false, a, false, bd, (short)0, accX[nt], false, false);
            accS[nt] = __builtin_amdgcn_wmma_f32_16x16x4_f32(
                false, s, false, bb, (short)0, accS[nt], false, false);
        }
    }

    int ncol = lane & 15;
    int mAdd = (lane >> 4) * 8;          // C/D layout: lanes>=16 hold rows M+8
#pragma unroll
    for (int nt = 0; nt < 4; ++nt) {
        int col = nt * 16 + ncol;
        float wc = Wc[col];
        float bb = bias[col];
#pragma unroll
        for (int v = 0; v < 8; ++v) {
            int node = node0 + v + mAdd;
            if (node >= N) continue;
            float g = gate[node];
            float val = g * (accX[nt][v] + bb) + invdeg[node] * accS[nt][v] + ssn[node] * wc;
            if (resid) val += resid[(size_t)node * HDIM + col];
            val = (val > 0.0f) ? val : 0.01f * val;   // leaky_relu, slope 0.01
            Y[(size_t)node * HDIM + col] = val;
        }
    }
}

// ------------------------------------------------- conv_out: H=64 -> 1, plus final +x residual
__global__ void conv_out_kernel(const float* __restrict__ X, const float* __restrict__ S,
                                const float* __restrict__ Wout, const float* __restrict__ bout,
                                const float* __restrict__ x0,
                                const float* __restrict__ invdeg, const float* __restrict__ gate,
                                const float* __restrict__ ssn, float* __restrict__ out, int N) {
    int n = blockIdx.x * blockDim.x + threadIdx.x;
    if (n >= N) return;
    float a1 = 0.0f, a2 = 0.0f;
    const float* Xr = X + (size_t)n * HDIM;
    const float* Sr = S + (size_t)n * HDIM;
#pragma unroll 8
    for (int k = 0; k < HDIM; ++k) {
        float wa = Wout[k], wb = Wout[HDIM + k];
        a1 += Xr[k] * (wa - wb);
        a2 += Sr[k] * wb;
    }
    out[n] = gate[n] * (a1 + bout[0]) + invdeg[n] * a2 + ssn[n] * Wout[2 * HDIM] + x0[n];
}

// ---------------------------------------------------------------- launcher
extern "C" void kernel_launch(void* const* d_in, const int* in_sizes, int n_in,
                              void* d_out, int out_size, void* d_ws, size_t ws_size,
                              hipStream_t stream) {
    const float* x    = (const float*)d_in[0];
    const int*   ei   = (const int*)  d_in[1];
    const float* Win  = (const float*)d_in[2];
    const float* bin  = (const float*)d_in[3];
    const float* Wmid = (const float*)d_in[4];
    const float* bmid = (const float*)d_in[5];
    const float* Wout = (const float*)d_in[6];
    const float* bout = (const float*)d_in[7];

    const int N = in_sizes[0];
    const int E = in_sizes[1] / 2;
    const int* src = ei;
    const int* dst = ei + E;

    float* w = (float*)d_ws;
    float* degacc  = w;
    float* signacc = w + (size_t)N;
    float* invdeg  = w + 2 * (size_t)N;
    float* gate    = w + 3 * (size_t)N;
    float* ssn     = w + 4 * (size_t)N;
    float* Sx      = w + 5 * (size_t)N;
    float* Wd6     = w + 6 * (size_t)N;
    float* bufA    = Wd6  + 6 * HDIM * HDIM;
    float* bufB    = bufA + (size_t)N * HDIM;
    float* bufC    = bufB + (size_t)N * HDIM;
    float* bufS    = bufC + (size_t)N * HDIM;

    const int nTiles = (N + 15) / 16;
    const int sWork  = E * 16;

    // node statistics + conv_in scalar neighbor sum
    zero_kernel<<<1024, 256, 0, stream>>>(w, 6 * N);
    edge_stats_kernel<<<(E + 255) / 256, 256, 0, stream>>>(src, dst, x, degacc, signacc, Sx, E);
    node_stats_kernel<<<(N + 255) / 256, 256, 0, stream>>>(degacc, signacc, invdeg, gate, ssn, N);
    wdiff_kernel<<<(6 * 4096 + 255) / 256, 256, 0, stream>>>(Wmid, Wd6, 6 * 4096);
    conv_in_kernel<<<(N * HDIM + 255) / 256, 256, 0, stream>>>(x, Win, bin, invdeg, gate, ssn, Sx,
                                                               bufA, N);

    float* cur = bufA;   // x2
    float* alt = bufC;
    for (int l = 0; l < 3; ++l) {
        // x1 = leaky_relu(conv(x2))
        zero_kernel<<<2048, 256, 0, stream>>>(bufS, N * HDIM);
        scatter_kernel<<<(sWork + 255) / 256, 256, 0, stream>>>(src, dst, cur, bufS, sWork);
        const float* Wl = Wmid + (size_t)(2 * l) * 129 * HDIM;
        mid_conv_kernel<<<(nTiles + 7) / 8, 256, 0, stream>>>(
            cur, bufS, Wd6 + (2 * l) * HDIM * HDIM, Wl + 64 * HDIM, Wl + 128 * HDIM,
            bmid + (2 * l) * HDIM, gate, invdeg, ssn, nullptr, bufB, nTiles, N);

        // x2 = leaky_relu(conv(x1) + x2)
        zero_kernel<<<2048, 256, 0, stream>>>(bufS, N * HDIM);
        scatter_kernel<<<(sWork + 255) / 256, 256, 0, stream>>>(src, dst, bufB, bufS, sWork);
        const float* Wl2 = Wmid + (size_t)(2 * l + 1) * 129 * HDIM;
        mid_conv_kernel<<<(nTiles + 7) / 8, 256, 0, stream>>>(
            bufB, bufS, Wd6 + (2 * l + 1) * HDIM * HDIM, Wl2 + 64 * HDIM, Wl2 + 128 * HDIM,
            bmid + (2 * l + 1) * HDIM, gate, invdeg, ssn, cur, alt, nTiles, N);

        float* t = cur; cur = alt; alt = t;
    }

    // out = conv_out(x2) + x
    zero_kernel<<<2048, 256, 0, stream>>>(bufS, N * HDIM);
    scatter_kernel<<<(sWork + 255) / 256, 256, 0, stream>>>(src, dst, cur, bufS, sWork);
    conv_out_kernel<<<(N + 255) / 256, 256, 0, stream>>>(cur, bufS, Wout, bout, x, invdeg, gate,
                                                         ssn, (float*)d_out, N);
}